// GATModel_1288490189679
// MI455X (gfx1250) — compile-verified
//
#include <hip/hip_runtime.h>
#include <math.h>

// ---------------------------------------------------------------------------
// GAT layer for MI455X (gfx1250, wave32).
// GEMMs use V_WMMA_F32_16X16X4_F32 (full fp32 precision; the op is scatter-
// bound so low-precision WMMA buys nothing). Edge phase uses fp32 atomics;
// h (51MB) is L2-resident on the 192MB L2.
// ---------------------------------------------------------------------------

typedef float v2f __attribute__((ext_vector_type(2)));
typedef float v8f __attribute__((ext_vector_type(8)));

#define HEADS  8
#define CDIM   32
#define HCDIM  256
#define NSLOPE 0.2f

// float atomic max via signed-int max / unsigned-int min bit trick.
// Requires destination initialized to -INF. Correct for all sign mixes.
__device__ __forceinline__ void atomicMaxF(float* addr, float val) {
    if (val >= 0.0f) {
        atomicMax((int*)addr, __float_as_int(val));
    } else {
        atomicMin((unsigned int*)addr, __float_as_uint(val));
    }
}

// ---------------------------------------------------------------------------
// Kernel 1: C[MxNB] = A[MxK] @ B[KxNB], fp32 WMMA, one wave per 16x16 tile.
// Lane layouts per CDNA5 ISA 7.12.2:
//   A 16x4 : lanes 0-15 hold M=lane,    VGPR0/1 = K {0,1}; lanes 16-31 = K {2,3}
//   B 4x16 : lanes 0-15 hold N=lane,    VGPR0/1 = K {0,1}; lanes 16-31 = K {2,3}
//   C 16x16: VGPR r = row r (lanes 0-15) / row 8+r (lanes 16-31), N = lane&15
// ---------------------------------------------------------------------------
__global__ void wmma_gemm_f32(const float* __restrict__ A,
                              const float* __restrict__ B,
                              float* __restrict__ C, int K, int NB) {
    const int  lane = threadIdx.x & 31;
    const bool hi   = lane >= 16;
    const int  l    = lane & 15;
    const int  row0 = blockIdx.x * 16;
    const int  col0 = blockIdx.y * 16;

    v8f acc = {};
    const float* arow = A + (size_t)(row0 + l) * K + (hi ? 2 : 0);
    const float* bcol = B + (size_t)(hi ? 2 : 0) * NB + col0 + l;
    for (int k = 0; k < K; k += 4) {
        v2f a; a.x = arow[k];                  a.y = arow[k + 1];
        v2f b; b.x = bcol[(size_t)k * NB];     b.y = bcol[(size_t)(k + 1) * NB];
        acc = __builtin_amdgcn_wmma_f32_16x16x4_f32(
                  false, a, false, b, (short)0, acc, false, false);
    }
    float* crow = C + (size_t)(row0 + (hi ? 8 : 0)) * NB + col0 + l;
#pragma unroll
    for (int r = 0; r < 8; ++r) crow[(size_t)r * NB] = acc[r];
}

// ---------------------------------------------------------------------------
// Kernel 7: out = relu(Acc + bias_gat) @ W_lin + b_lin. Same WMMA scheme,
// bias+relu fused into the A-operand load, b_lin in the epilogue.
// ---------------------------------------------------------------------------
__global__ void wmma_gemm_relu_bias(const float* __restrict__ Acc,
                                    const float* __restrict__ bias,
                                    const float* __restrict__ B,
                                    const float* __restrict__ bout,
                                    float* __restrict__ C, int K, int NB) {
    const int  lane = threadIdx.x & 31;
    const bool hi   = lane >= 16;
    const int  l    = lane & 15;
    const int  row0 = blockIdx.x * 16;
    const int  col0 = blockIdx.y * 16;

    v8f acc = {};
    const float* arow = Acc + (size_t)(row0 + l) * K;
    const float* bcol = B + (size_t)(hi ? 2 : 0) * NB + col0 + l;
    const int    koff = hi ? 2 : 0;
    for (int k = 0; k < K; k += 4) {
        const int kk = k + koff;
        v2f a;
        a.x = fmaxf(arow[kk]     + bias[kk],     0.0f);
        a.y = fmaxf(arow[kk + 1] + bias[kk + 1], 0.0f);
        v2f b; b.x = bcol[(size_t)k * NB]; b.y = bcol[(size_t)(k + 1) * NB];
        acc = __builtin_amdgcn_wmma_f32_16x16x4_f32(
                  false, a, false, b, (short)0, acc, false, false);
    }
    const float bo = bout[col0 + l];
    float* crow = C + (size_t)(row0 + (hi ? 8 : 0)) * NB + col0 + l;
#pragma unroll
    for (int r = 0; r < 8; ++r) crow[(size_t)r * NB] = acc[r] + bo;
}

// ---------------------------------------------------------------------------
// Kernel 2: per-(node,head) attention scores + init of segment buffers.
// ---------------------------------------------------------------------------
__global__ void gat_att_scores(const float* __restrict__ h,
                               const float* __restrict__ att_src,
                               const float* __restrict__ att_dst,
                               float* __restrict__ a_src,
                               float* __restrict__ a_dst,
                               float* __restrict__ maxbuf,
                               float* __restrict__ denom, int N) {
    const int t = blockIdx.x * blockDim.x + threadIdx.x;
    if (t >= N * HEADS) return;
    const int n  = t / HEADS;
    const int hd = t - n * HEADS;
    const float* hp = h + (size_t)n * HCDIM + hd * CDIM;
    const float* as = att_src + hd * CDIM;
    const float* ad = att_dst + hd * CDIM;
    float s0 = 0.0f, s1 = 0.0f;
#pragma unroll
    for (int c = 0; c < CDIM; ++c) {
        const float v = hp[c];
        s0 = fmaf(v, as[c], s0);
        s1 = fmaf(v, ad[c], s1);
    }
    a_src[t]  = s0;
    a_dst[t]  = s1;
    maxbuf[t] = -INFINITY;
    denom[t]  = 0.0f;
}

__global__ void zero_f32(float* __restrict__ p, int n) {
    const int t = blockIdx.x * blockDim.x + threadIdx.x;
    if (t < n) p[t] = 0.0f;
}

// edge helpers: edges [0,E) from edge_index, edges [E, E+N) are self-loops.
__device__ __forceinline__ void edge_src_dst(const int* __restrict__ ei,
                                             int e, int E, int& s, int& d) {
    if (e < E) { s = ei[e]; d = ei[E + e]; }
    else       { s = e - E; d = e - E; }
}

__device__ __forceinline__ float edge_logit(const float* __restrict__ a_src,
                                            const float* __restrict__ a_dst,
                                            int s, int d, int hd) {
    float v = a_src[s * HEADS + hd] + a_dst[d * HEADS + hd];
    return v > 0.0f ? v : NSLOPE * v;   // leaky_relu
}

// ---------------------------------------------------------------------------
// Kernel 4: segment max of edge logits over dst.
// ---------------------------------------------------------------------------
__global__ void gat_edge_max(const int* __restrict__ ei,
                             const float* __restrict__ a_src,
                             const float* __restrict__ a_dst,
                             float* __restrict__ maxbuf, int E, int ET) {
    const int t = blockIdx.x * blockDim.x + threadIdx.x;
    if (t >= ET * HEADS) return;
    const int e  = t >> 3;
    const int hd = t & 7;
    int s, d;
    edge_src_dst(ei, e, E, s, d);
    atomicMaxF(&maxbuf[d * HEADS + hd], edge_logit(a_src, a_dst, s, d, hd));
}

// ---------------------------------------------------------------------------
// Kernel 5: denom += exp(e - max). Logit recomputed (cheaper than a 27MB buffer).
// ---------------------------------------------------------------------------
__global__ void gat_edge_expsum(const int* __restrict__ ei,
                                const float* __restrict__ a_src,
                                const float* __restrict__ a_dst,
                                const float* __restrict__ maxbuf,
                                float* __restrict__ denom, int E, int ET) {
    const int t = blockIdx.x * blockDim.x + threadIdx.x;
    if (t >= ET * HEADS) return;
    const int e  = t >> 3;
    const int hd = t & 7;
    int s, d;
    edge_src_dst(ei, e, E, s, d);
    const float v  = edge_logit(a_src, a_dst, s, d, hd);
    const float ex = expf(v - maxbuf[d * HEADS + hd]);
    atomicAdd(&denom[d * HEADS + hd], ex);
}

// ---------------------------------------------------------------------------
// Kernel 6: scatter messages. One wave32 per edge; lanes compute the 8 head
// alphas redundantly (h = lane&7), broadcast per-head alpha via __shfl, then
// each lane atomically accumulates 8 of the 256 channels.
// ---------------------------------------------------------------------------
__global__ void gat_edge_scatter(const int* __restrict__ ei,
                                 const float* __restrict__ a_src,
                                 const float* __restrict__ a_dst,
                                 const float* __restrict__ maxbuf,
                                 const float* __restrict__ denom,
                                 const float* __restrict__ h,
                                 float* __restrict__ accum, int E, int ET) {
    const int lane = threadIdx.x & 31;
    const int wid  = (blockIdx.x * blockDim.x + threadIdx.x) >> 5;
    if (wid >= ET) return;
    int s, d;
    edge_src_dst(ei, wid, E, s, d);

    const int   h8    = lane & 7;
    const float v     = edge_logit(a_src, a_dst, s, d, h8);
    const float ex    = expf(v - maxbuf[d * HEADS + h8]);
    const float alpha = ex / (denom[d * HEADS + h8] + 1e-16f);

    const float* hs = h + (size_t)s * HCDIM;
    float*       ap = accum + (size_t)d * HCDIM;
#pragma unroll
    for (int i = 0; i < HEADS; ++i) {
        const float al = __shfl(alpha, i, 32);       // head i alpha
        atomicAdd(&ap[i * CDIM + lane], hs[i * CDIM + lane] * al);
    }
}

// ---------------------------------------------------------------------------
extern "C" void kernel_launch(void* const* d_in, const int* in_sizes, int n_in,
                              void* d_out, int out_size, void* d_ws, size_t ws_size,
                              hipStream_t stream) {
    (void)n_in; (void)out_size; (void)ws_size;

    const float* x        = (const float*)d_in[0];
    const int*   ei       = (const int*)d_in[1];   // [2, E] int32
    const float* W_gat    = (const float*)d_in[2];
    const float* att_src  = (const float*)d_in[3];
    const float* att_dst  = (const float*)d_in[4];
    const float* bias_gat = (const float*)d_in[5];
    const float* W_lin    = (const float*)d_in[6];
    const float* b_lin    = (const float*)d_in[7];
    float*       out      = (float*)d_out;

    const int IN  = 256;
    const int N   = in_sizes[0] / IN;        // 50000 (multiple of 16)
    const int E   = in_sizes[1] / 2;         // 800000
    const int OUT = in_sizes[6] / HCDIM;     // 64
    const int ET  = E + N;                   // with self-loops

    // workspace layout (fp32): h | accum | a_src | a_dst | maxbuf | denom
    float* ws   = (float*)d_ws;
    float* h    = ws;
    float* acc  = h    + (size_t)N * HCDIM;
    float* asrc = acc  + (size_t)N * HCDIM;
    float* adst = asrc + (size_t)N * HEADS;
    float* mbuf = adst + (size_t)N * HEADS;
    float* den  = mbuf + (size_t)N * HEADS;

    // 1. h = x @ W_gat   (fp32 WMMA)
    wmma_gemm_f32<<<dim3(N / 16, HCDIM / 16), 32, 0, stream>>>(x, W_gat, h, IN, HCDIM);

    // 2. attention scores + buffer init
    const int t2 = N * HEADS;
    gat_att_scores<<<(t2 + 255) / 256, 256, 0, stream>>>(h, att_src, att_dst,
                                                         asrc, adst, mbuf, den, N);

    // 3. zero message accumulator
    const int t3 = N * HCDIM;
    zero_f32<<<(t3 + 255) / 256, 256, 0, stream>>>(acc, t3);

    // 4. segment max
    const int t4 = ET * HEADS;
    gat_edge_max<<<(t4 + 255) / 256, 256, 0, stream>>>(ei, asrc, adst, mbuf, E, ET);

    // 5. segment exp-sum
    gat_edge_expsum<<<(t4 + 255) / 256, 256, 0, stream>>>(ei, asrc, adst, mbuf, den, E, ET);

    // 6. weighted message scatter (one wave per edge, 8 waves / block)
    gat_edge_scatter<<<(ET + 7) / 8, 256, 0, stream>>>(ei, asrc, adst, mbuf, den,
                                                       h, acc, E, ET);

    // 7. out = relu(acc + bias_gat) @ W_lin + b_lin   (fused fp32 WMMA)
    wmma_gemm_relu_bias<<<dim3(N / 16, OUT / 16), 32, 0, stream>>>(acc, bias_gat,
                                                                   W_lin, b_lin,
                                                                   out, HCDIM, OUT);
}